// QuantizeLayer_52664888984101
// MI455X (gfx1250) — compile-verified
//
#include <hip/hip_runtime.h>
#include <stdint.h>

// Per-channel fake-quant (quantize-dequantize) of x[64,256,56,56] fp32.
//   out = clip(round(x / s_c), -128, 127) * s_c
// Memory-bound: ~411 MB traffic -> ~17.6 us floor at 23.3 TB/s.
// Strategy: one 256-thread block per (b,c) plane (784 float4, contiguous),
// wave-uniform scale via scalar load + single divide, double-buffered
// CDNA5 async global->LDS loads (ASYNCcnt), NT 128-bit stores.

typedef float v4f __attribute__((ext_vector_type(4)));

#define NCHAN   256
#define PLANES  (64 * NCHAN)   // 16384 (b,c) planes
#define PLANE4  784            // 56*56/4 float4 per plane
#define TPB     256
#define CHUNKS  4              // ceil(784/256)

__global__ __launch_bounds__(TPB)
void fakequant_kernel(const float* __restrict__ x,
                      const float* __restrict__ scale,
                      float* __restrict__ out)
{
    __shared__ v4f lds[2][TPB];

    const uint32_t tid   = threadIdx.x;
    const uint32_t plane = blockIdx.x;          // plane = b*256 + c
    const uint32_t c     = plane & (NCHAN - 1); // c = plane % 256 (uniform)

    const float s    = scale[c];                // uniform -> s_load
    const float rinv = 1.0f / s;                // one IEEE divide per block

    const float* xbase = x   + (size_t)plane * (PLANE4 * 4);
    float*       obase = out + (size_t)plane * (PLANE4 * 4);

    // Low 32 bits of a flat shared pointer == LDS byte offset (aperture is in
    // the high dword on gfx1250), which is exactly what the async-to-LDS
    // instruction's VDST operand wants.
    const uint32_t lds_off0 = (uint32_t)(uintptr_t)&lds[0][tid];
    const uint32_t lds_off1 = (uint32_t)(uintptr_t)&lds[1][tid];

    // Clamp the float4 index so EVERY lane of EVERY wave issues each chunk's
    // async load (keeps per-wave ASYNCcnt uniform; duplicate loads hit cache).
    auto clampi = [](uint32_t i) { return i < (uint32_t)PLANE4 ? i : (uint32_t)(PLANE4 - 1); };

    // Prologue: issue chunk 0 into buffer 0.
    {
        const uint32_t goff = clampi(tid) * 16u;
        asm volatile("global_load_async_to_lds_b128 %0, %1, %2"
                     :: "v"(lds_off0), "v"(goff), "s"(xbase)
                     : "memory");
    }

    #pragma unroll
    for (int k = 0; k < CHUNKS; ++k) {
        if (k + 1 < CHUNKS) {
            // Issue next chunk into the other buffer, then wait for current.
            const uint32_t goff = clampi((uint32_t)(k + 1) * TPB + tid) * 16u;
            const uint32_t ldst = ((k + 1) & 1) ? lds_off1 : lds_off0;
            asm volatile("global_load_async_to_lds_b128 %0, %1, %2"
                         :: "v"(ldst), "v"(goff), "s"(xbase)
                         : "memory");
            asm volatile("s_wait_asynccnt 1" ::: "memory");
        } else {
            asm volatile("s_wait_asynccnt 0" ::: "memory");
        }

        // Each lane reads back its own 16B slot -> no workgroup barrier needed.
        v4f v = lds[k & 1][tid];

        v4f q;
        #pragma unroll
        for (int j = 0; j < 4; ++j) {
            float t = rintf(v[j] * rinv);            // v_rndne_f32 (round half to even)
            t = fminf(fmaxf(t, -128.0f), 127.0f);    // -> v_med3_f32
            q[j] = t * s;
        }

        const uint32_t idx = (uint32_t)k * TPB + tid;
        if (idx < PLANE4) {
            __builtin_nontemporal_store(q, (v4f*)(obase + (size_t)idx * 4));
        }
    }
}

extern "C" void kernel_launch(void* const* d_in, const int* in_sizes, int n_in,
                              void* d_out, int out_size, void* d_ws, size_t ws_size,
                              hipStream_t stream) {
    (void)in_sizes; (void)n_in; (void)d_ws; (void)ws_size; (void)out_size;
    const float* x     = (const float*)d_in[0];   // [64,256,56,56] fp32
    const float* scale = (const float*)d_in[1];   // [256] fp32
    float*       out   = (float*)d_out;           // [64,256,56,56] fp32

    fakequant_kernel<<<PLANES, TPB, 0, stream>>>(x, scale, out);
}